// CPL_Loss_72799695667404
// MI455X (gfx1250) — compile-verified
//
#include <hip/hip_runtime.h>
#include <hip/hip_bf16.h>
#include <math.h>

// Problem constants (match reference).
#define N_WAY     128
#define PER_CLASS 128
#define QN        (N_WAY * PER_CLASS)   // 16384 queries
#define DIM       1024
#define NNEG      ((N_WAY - 1) * 4)     // 508 negatives per query
#define EPS_F     1e-8f

typedef __attribute__((ext_vector_type(2))) float v2f;
typedef __attribute__((ext_vector_type(8))) float v8f;

__device__ __forceinline__ float wave_sum(float v) {
#pragma unroll
  for (int off = 16; off > 0; off >>= 1) v += __shfl_xor(v, off, 32);
  return v;
}

// ---------------------------------------------------------------------------
// Kernel 1: normalize prototype rows into pn.  One wave32 per row.
// ---------------------------------------------------------------------------
__global__ void proto_norm_kernel(const float* __restrict__ proto,
                                  float* __restrict__ pn) {
  const int wave = (blockIdx.x * blockDim.x + threadIdx.x) >> 5;
  const int lane = threadIdx.x & 31;
  if (wave >= N_WAY) return;
  const float* src = proto + (size_t)wave * DIM;
  float s = 0.f;
  for (int j = lane; j < DIM; j += 32) { float v = src[j]; s += v * v; }
  s = wave_sum(s);
  const float inv = 1.f / fmaxf(sqrtf(s), EPS_F);
  float* dst = pn + (size_t)wave * DIM;
  for (int j = lane; j < DIM; j += 32) dst[j] = src[j] * inv;
}

// ---------------------------------------------------------------------------
// Kernel 2: per-query inverse norms.  One wave32 per query row.
// ---------------------------------------------------------------------------
__global__ void query_invnorm_kernel(const float* __restrict__ queries,
                                     float* __restrict__ invq) {
  const int wave = (blockIdx.x * blockDim.x + threadIdx.x) >> 5;
  const int lane = threadIdx.x & 31;
  if (wave >= QN) return;
  const float* src = queries + (size_t)wave * DIM;
  float s = 0.f;
  for (int j = lane; j < DIM; j += 32) { float v = src[j]; s += v * v; }
  s = wave_sum(s);
  if (lane == 0) invq[wave] = 1.f / fmaxf(sqrtf(s), EPS_F);
}

// ---------------------------------------------------------------------------
// Kernel 3: S[p][q] = pn[p] . queries[q] * invq[q]  via V_WMMA_F32_16X16X4_F32.
// One wave owns a 16-query strip x all 128 prototypes (8 accumulator tiles),
// so queries stream from HBM exactly once.
//
// A-matrix 16x4 f32 lane layout (ISA 7.12.2): lane L holds M = L&15,
// K = 2*(L>>4) + {0,1}  -> contiguous float2 from a row-major matrix.
// B-matrix 4x16: lane L holds N = L&15, K = 2*(L>>4) + {0,1} -> same
// addressing pattern from row-major queries.
// C/D 16x16 f32: VGPR v, lane L -> row M = v + 8*(L>>4), col N = L&15.
// ---------------------------------------------------------------------------
__global__ void __launch_bounds__(256)
sim_wmma_kernel(const float* __restrict__ queries,
                const float* __restrict__ pn,
                const float* __restrict__ invq,
                float* __restrict__ S) {
  const int gw   = (blockIdx.x * blockDim.x + threadIdx.x) >> 5;  // 0..1023
  const int lane = threadIdx.x & 31;
  const int lo   = lane & 15;
  const int hi   = lane >> 4;
  const int q0   = gw * 16;

  const float* qrow = queries + (size_t)(q0 + lo) * DIM + 2 * hi;
  const float* prow = pn + (size_t)lo * DIM + 2 * hi;

  v8f acc[8] = {};

  for (int kc = 0; kc < DIM; kc += 4) {
    v2f b = *(const v2f*)(qrow + kc);
    __builtin_prefetch(qrow + kc + 128, 0, 0);  // global_prefetch_b8
#pragma unroll
    for (int t = 0; t < 8; ++t) {
      v2f a = *(const v2f*)(prow + (size_t)t * 16 * DIM + kc);
      // 8 args: (neg_a, A, neg_b, B, c_mod, C, reuse_a, reuse_b)
      acc[t] = __builtin_amdgcn_wmma_f32_16x16x4_f32(
          false, a, false, b, (short)0, acc[t], false, false);
    }
  }

  const float inv = invq[q0 + lo];  // this lane's column scale
#pragma unroll
  for (int t = 0; t < 8; ++t) {
#pragma unroll
    for (int v = 0; v < 8; ++v) {
      const int p = t * 16 + v + 8 * hi;
      S[(size_t)p * QN + q0 + lo] = acc[t][v] * inv;
    }
  }
}

// ---------------------------------------------------------------------------
// Kernel 4a: zero the scalar output (harness poisons it).
// ---------------------------------------------------------------------------
__global__ void zero_out_kernel(float* __restrict__ out) {
  if (threadIdx.x == 0 && blockIdx.x == 0) out[0] = 0.f;
}

// ---------------------------------------------------------------------------
// Kernel 4b: per-query online logsumexp over 508 gathered negatives + positive,
// one wave32 per query; single atomicAdd of the loss contribution.
// ---------------------------------------------------------------------------
__global__ void __launch_bounds__(256)
loss_kernel(const float* __restrict__ S,
            const int* __restrict__ labels,
            const int* __restrict__ neg_idx,
            float* __restrict__ out) {
  const int q    = (blockIdx.x * blockDim.x + threadIdx.x) >> 5;
  const int lane = threadIdx.x & 31;
  if (q >= QN) return;

  const int lab = labels[q];
  const float* Srow = S + (size_t)lab * QN;
  const int* nidx   = neg_idx + (size_t)q * NNEG;

  // Online logsumexp accumulation per lane.
  float m = -1e30f, s = 0.f;
  for (int i = lane; i < NNEG; i += 32) {
    const float v = Srow[nidx[i]];
    if (v > m) { s = s * __expf(m - v) + 1.f; m = v; }
    else       { s += __expf(v - m); }
  }
  const float pos = Srow[q];
  if (lane == 0) {  // fold the positive logit into lane 0's state
    if (pos > m) { s = s * __expf(m - pos) + 1.f; m = pos; }
    else         { s += __expf(pos - m); }
  }

  // Cross-lane merge of (max, scaled-sum) pairs.
#pragma unroll
  for (int off = 16; off > 0; off >>= 1) {
    const float mo = __shfl_xor(m, off, 32);
    const float so = __shfl_xor(s, off, 32);
    const float mn = fmaxf(m, mo);
    s = s * __expf(m - mn) + so * __expf(mo - mn);
    m = mn;
  }

  if (lane == 0) {
    const float lse = m + __logf(s);
    // loss = -mean(logp_last) / Q  ->  sum over q of -(pos - lse) / Q^2
    const float scale = 1.f / ((float)QN * (float)QN);
    atomicAdd(out, -(pos - lse) * scale);
  }
}

// ---------------------------------------------------------------------------
extern "C" void kernel_launch(void* const* d_in, const int* in_sizes, int n_in,
                              void* d_out, int out_size, void* d_ws, size_t ws_size,
                              hipStream_t stream) {
  (void)in_sizes; (void)n_in; (void)out_size; (void)ws_size;

  const float* proto   = (const float*)d_in[0];  // 128*1024 f32
  const float* queries = (const float*)d_in[1];  // 16384*1024 f32
  const int*   labels  = (const int*)d_in[2];    // 16384 i32
  const int*   neg_idx = (const int*)d_in[3];    // 16384*508 i32
  float*       out     = (float*)d_out;          // scalar f32

  // Workspace layout (floats): pn[128*1024] | invq[16384] | S[128*16384]
  float* pn   = (float*)d_ws;
  float* invq = pn + (size_t)N_WAY * DIM;
  float* S    = invq + QN;

  // 1) prototype normalization: 128 waves (block=128 -> 4 waves, 32 blocks)
  proto_norm_kernel<<<32, 128, 0, stream>>>(proto, pn);

  // 2) query inverse norms: 16384 waves (block=256 -> 8 waves, 2048 blocks)
  query_invnorm_kernel<<<2048, 256, 0, stream>>>(queries, invq);

  // 3) WMMA similarity matrix: 1024 waves (block=256 -> 8 waves, 128 blocks)
  sim_wmma_kernel<<<128, 256, 0, stream>>>(queries, pn, invq, S);

  // 4) loss reduction
  zero_out_kernel<<<1, 32, 0, stream>>>(out);
  loss_kernel<<<2048, 256, 0, stream>>>(S, labels, neg_idx, out);
}